// EnhancedDGDNModel_28286654611706
// MI455X (gfx1250) — compile-verified
//
#include <hip/hip_runtime.h>
#include <hip/hip_bf16.h>
#include <cstddef>

typedef __attribute__((ext_vector_type(2))) float v2f;
typedef __attribute__((ext_vector_type(8))) float v8f;

#define NN   2048      // nodes
#define NE   131072    // edges
#define ND   64        // node dim
#define HID  256
#define NH   8
#define HD   32        // head dim
#define LN_EPS 1e-5f

// ---------------------------------------------------------------------------
// fp32 WMMA: D(16x16) += A(16x4) x B(4x16).  A-frag: lane l holds row M=l%16,
// elems j -> K = 2*(l/16)+j.  B-frag mirrored.  C/D: lane l vgpr r holds
// (M = r + 8*(l/16), N = l%16).
// ---------------------------------------------------------------------------
__device__ __forceinline__ v8f wmma4(v2f a, v2f b, v8f c) {
  return __builtin_amdgcn_wmma_f32_16x16x4_f32(false, a, false, b, (short)0, c,
                                               false, false);
}

__device__ __forceinline__ void store_tile(float* __restrict__ C, v8f acc,
                                           const float* __restrict__ bias,
                                           int m0, int col, int N, int relu,
                                           int hi) {
  float bv = bias ? bias[col] : 0.f;
  #pragma unroll
  for (int r = 0; r < 8; ++r) {
    float v = acc[r] + bv;
    if (relu) v = fmaxf(v, 0.f);
    C[(size_t)(m0 + r + 8 * hi) * N + col] = v;
  }
}

// ---------------------------------------------------------------------------
// Register-blocked GEMM: each wave computes a 32x64 tile of
// C[M,N] = act(A[M,K] @ W[N,K]^T + bias).  2 A-frags x 4 B-frags feed 8 WMMAs
// per K-step (~2.7x higher L2 arithmetic intensity than 16x16/wave).
// Requires M%32==0, N%64==0, K%4==0 (grid covers exactly; EXEC all-ones).
// ---------------------------------------------------------------------------
__global__ void gemm_wt_t24_kernel(const float* __restrict__ A,
                                   const float* __restrict__ W,
                                   const float* __restrict__ bias,
                                   float* __restrict__ C,
                                   int M, int N, int K, int relu) {
  const int lane = threadIdx.x;
  const int m16  = lane & 15;
  const int hi   = lane >> 4;
  const int tn   = blockIdx.x, tm = blockIdx.y;

  const float* ap0 = A + (size_t)(tm * 32 + m16) * K + hi * 2;
  const float* ap1 = ap0 + (size_t)16 * K;
  const float* wp0 = W + (size_t)(tn * 64 + m16) * K + hi * 2;
  const float* wp1 = wp0 + (size_t)16 * K;
  const float* wp2 = wp0 + (size_t)32 * K;
  const float* wp3 = wp0 + (size_t)48 * K;

  v8f a00 = {}, a01 = {}, a02 = {}, a03 = {};
  v8f a10 = {}, a11 = {}, a12 = {}, a13 = {};
  for (int k = 0; k < K; k += 4) {
    v2f fa0 = { ap0[k], ap0[k + 1] };
    v2f fa1 = { ap1[k], ap1[k + 1] };
    v2f fb0 = { wp0[k], wp0[k + 1] };
    v2f fb1 = { wp1[k], wp1[k + 1] };
    v2f fb2 = { wp2[k], wp2[k + 1] };
    v2f fb3 = { wp3[k], wp3[k + 1] };
    a00 = wmma4(fa0, fb0, a00);
    a01 = wmma4(fa0, fb1, a01);
    a02 = wmma4(fa0, fb2, a02);
    a03 = wmma4(fa0, fb3, a03);
    a10 = wmma4(fa1, fb0, a10);
    a11 = wmma4(fa1, fb1, a11);
    a12 = wmma4(fa1, fb2, a12);
    a13 = wmma4(fa1, fb3, a13);
  }
  const int m0 = tm * 32, c0 = tn * 64 + m16;
  store_tile(C, a00, bias, m0,      c0,      N, relu, hi);
  store_tile(C, a01, bias, m0,      c0 + 16, N, relu, hi);
  store_tile(C, a02, bias, m0,      c0 + 32, N, relu, hi);
  store_tile(C, a03, bias, m0,      c0 + 48, N, relu, hi);
  store_tile(C, a10, bias, m0 + 16, c0,      N, relu, hi);
  store_tile(C, a11, bias, m0 + 16, c0 + 16, N, relu, hi);
  store_tile(C, a12, bias, m0 + 16, c0 + 32, N, relu, hi);
  store_tile(C, a13, bias, m0 + 16, c0 + 48, N, relu, hi);
}

// Fallback GEMM (irregular N, e.g. the N=2 uncertainty head): one wave per
// 16x16 tile; OOB rows clamped (dup compute, masked store) so EXEC stays 1s.
__global__ void gemm_wt_kernel(const float* __restrict__ A,
                               const float* __restrict__ W,
                               const float* __restrict__ bias,
                               float* __restrict__ C,
                               int M, int N, int K, int relu) {
  const int lane = threadIdx.x;
  const int m16  = lane & 15;
  const int hi   = lane >> 4;
  const int tn   = blockIdx.x, tm = blockIdx.y;

  int arow = tm * 16 + m16;  arow = arow < M ? arow : M - 1;
  int wrow = tn * 16 + m16;  wrow = wrow < N ? wrow : N - 1;
  const float* ap = A + (size_t)arow * K + hi * 2;
  const float* wp = W + (size_t)wrow * K + hi * 2;

  v8f acc = {};
  for (int k = 0; k < K; k += 4) {
    v2f a = { ap[k], ap[k + 1] };
    v2f b = { wp[k], wp[k + 1] };
    acc = wmma4(a, b, acc);
  }

  const int n = tn * 16 + m16;
  #pragma unroll
  for (int r = 0; r < 8; ++r) {
    int m = tm * 16 + r + 8 * hi;
    if (m < M && n < N) {
      float v = acc[r];
      if (bias) v += bias[n];
      if (relu) v = fmaxf(v, 0.f);
      C[(size_t)m * N + n] = v;
    }
  }
}

// ---------------------------------------------------------------------------
// Per-edge time MLP (1->32->64) + scatter of the 64-dim hidden.
// segsum(relu2 @ w3^T) = segsum(relu2) @ w3^T + cnt*b3 (linearity) => we only
// scatter 64 dims here; the w3 GEMM runs once per *node* afterwards.
// ---------------------------------------------------------------------------
__global__ void edge_time_mlp_kernel(const float* __restrict__ ts,
                                     const int* __restrict__ src,
                                     const int* __restrict__ dst,
                                     const float* __restrict__ w1,
                                     const float* __restrict__ b1,
                                     const float* __restrict__ w2,
                                     const float* __restrict__ b2,
                                     float* __restrict__ S64,
                                     float inv_scale) {
  __shared__ float sh1[32];
  const int e = blockIdx.x;
  const int j = threadIdx.x;             // 0..63
  const float t = ts[e] * inv_scale;
  if (j < 32) sh1[j] = fmaxf(fmaf(w1[j], t, b1[j]), 0.f);
  __syncthreads();
  float acc = b2[j];
  const float* w2r = w2 + j * 32;
  #pragma unroll
  for (int i = 0; i < 32; ++i) acc = fmaf(w2r[i], sh1[i], acc);
  const float h2 = fmaxf(acc, 0.f);
  const int s = src[e], d = dst[e];
  atomicAdd(&S64[(size_t)s * 64 + j], h2);
  if (d != s) atomicAdd(&S64[(size_t)d * 64 + j], h2);
}

__global__ void edge_cnt_kernel(const int* __restrict__ src,
                                const int* __restrict__ dst,
                                float* __restrict__ cnt) {
  int e = blockIdx.x * blockDim.x + threadIdx.x;
  if (e >= NE) return;
  int s = src[e], d = dst[e];
  atomicAdd(&cnt[s], 1.f);
  if (d != s) atomicAdd(&cnt[d], 1.f);
}

// h0 = x + (cnt>0 ? T/cnt + b3 : 0)
__global__ void ntf_combine_kernel(const float* __restrict__ x,
                                   const float* __restrict__ T,
                                   const float* __restrict__ cnt,
                                   const float* __restrict__ b3,
                                   float* __restrict__ h0) {
  int i = blockIdx.x * blockDim.x + threadIdx.x;
  if (i >= NN * HID) return;
  int n = i >> 8, c = i & 255;
  float ct = cnt[n];
  float v = (ct > 0.f) ? (T[i] / ct + b3[c]) : 0.f;
  h0[i] = x[i] + v;
}

// ---------------------------------------------------------------------------
// Flash attention, S=2048, hd=32, fp32 WMMA.  One wave per (16-row qblock,
// head).  Q held in registers across all 128 key tiles; online softmax; P
// staged through LDS to relayout C-frag -> A-frag.
// ---------------------------------------------------------------------------
__global__ void flash_attn_kernel(const float* __restrict__ qkv,  // [S,768]
                                  float* __restrict__ out) {       // [S,256]
  __shared__ float pls[16][17];
  const int lane = threadIdx.x;
  const int m16  = lane & 15;
  const int hi   = lane >> 4;
  const int qb   = blockIdx.x;   // S/16
  const int head = blockIdx.y;
  const int qoff = head * HD, koff = HID + head * HD, voff = 2 * HID + head * HD;
  const float sc = 0.17677669529663687f;   // 1/sqrt(32)

  const int qrow = qb * 16 + m16;
  float qa[8][2];
  #pragma unroll
  for (int st = 0; st < 8; ++st) {
    const float* qp = qkv + (size_t)qrow * 768 + qoff + st * 4 + hi * 2;
    qa[st][0] = qp[0]; qa[st][1] = qp[1];
  }

  v8f o0 = {}, o1 = {};
  float rmax[8], rsum[8];
  #pragma unroll
  for (int r = 0; r < 8; ++r) { rmax[r] = -3.0e38f; rsum[r] = 0.f; }

  for (int kt = 0; kt < NN / 16; ++kt) {
    v8f s = {};
    #pragma unroll
    for (int st = 0; st < 8; ++st) {
      v2f a = { qa[st][0], qa[st][1] };
      const float* kp = qkv + (size_t)(kt * 16 + m16) * 768 + koff + st * 4 + hi * 2;
      v2f b = { kp[0], kp[1] };
      s = wmma4(a, b, s);
    }
    float p[8];
    #pragma unroll
    for (int r = 0; r < 8; ++r) {
      float v = s[r] * sc;
      float mx = v;
      for (int off = 1; off < 16; off <<= 1) mx = fmaxf(mx, __shfl_xor(mx, off, 16));
      float nm = fmaxf(rmax[r], mx);
      float corr = __expf(rmax[r] - nm);
      float pv = __expf(v - nm);
      float ps = pv;
      for (int off = 1; off < 16; off <<= 1) ps += __shfl_xor(ps, off, 16);
      rsum[r] = rsum[r] * corr + ps;
      rmax[r] = nm;
      o0[r] *= corr; o1[r] *= corr;
      p[r] = pv;
    }
    __syncthreads();
    #pragma unroll
    for (int r = 0; r < 8; ++r) pls[r + 8 * hi][m16] = p[r];
    __syncthreads();
    #pragma unroll
    for (int st = 0; st < 4; ++st) {
      v2f a = { pls[m16][st * 4 + hi * 2], pls[m16][st * 4 + hi * 2 + 1] };
      const float* vp = qkv + (size_t)(kt * 16 + st * 4 + hi * 2) * 768 + voff;
      v2f b0 = { vp[m16],      vp[768 + m16] };
      v2f b1v = { vp[m16 + 16], vp[768 + m16 + 16] };
      o0 = wmma4(a, b0, o0);
      o1 = wmma4(a, b1v, o1);
    }
  }
  #pragma unroll
  for (int r = 0; r < 8; ++r) {
    float inv = 1.f / rsum[r];
    int row = qb * 16 + r + 8 * hi;
    out[(size_t)row * HID + head * HD + m16]      = o0[r] * inv;
    out[(size_t)row * HID + head * HD + 16 + m16] = o1[r] * inv;
  }
}

// out = LN(x + y) * gamma + beta    (one 256-thread block per row)
__global__ void add_ln_kernel(const float* __restrict__ x,
                              const float* __restrict__ y,
                              const float* __restrict__ gamma,
                              const float* __restrict__ beta,
                              float* __restrict__ out) {
  __shared__ float red[256];
  const int row = blockIdx.x, t = threadIdx.x;
  const size_t i = (size_t)row * HID + t;
  float v = x[i] + y[i];
  red[t] = v; __syncthreads();
  for (int s = 128; s > 0; s >>= 1) { if (t < s) red[t] += red[t + s]; __syncthreads(); }
  float mean = red[0] * (1.f / HID);
  __syncthreads();
  float d = v - mean;
  red[t] = d * d; __syncthreads();
  for (int s = 128; s > 0; s >>= 1) { if (t < s) red[t] += red[t + s]; __syncthreads(); }
  float var = red[0] * (1.f / HID);
  out[i] = (v - mean) * rsqrtf(var + LN_EPS) * gamma[t] + beta[t];
}

__global__ void rownorm_acc_kernel(const float* __restrict__ h, float* acc) {
  __shared__ float red[256];
  const int row = blockIdx.x, t = threadIdx.x;
  float v = h[(size_t)row * HID + t];
  red[t] = v * v; __syncthreads();
  for (int s = 128; s > 0; s >>= 1) { if (t < s) red[t] += red[t + s]; __syncthreads(); }
  if (t == 0) atomicAdd(acc, sqrtf(red[0]));
}

__global__ void interleave3_kernel(const float* e0, const float* e1,
                                   const float* e2, float* stacked) {
  int i = blockIdx.x * blockDim.x + threadIdx.x;
  if (i >= NN * HID) return;
  int n = i >> 8, c = i & 255;
  stacked[(size_t)(n * 3 + 0) * HID + c] = e0[i];
  stacked[(size_t)(n * 3 + 1) * HID + c] = e1[i];
  stacked[(size_t)(n * 3 + 2) * HID + c] = e2[i];
}

// fusion MHA over S=3: one thread per (node, head); also accumulates
// head-averaged attention into d_out (pre-zeroed).
__global__ void fusion_attn_kernel(const float* __restrict__ qkv,  // [6144,768]
                                   float* __restrict__ aout,       // [6144,256]
                                   float* __restrict__ attnw) {    // [2048,3,3]
  int idx = blockIdx.x * blockDim.x + threadIdx.x;
  if (idx >= NN * NH) return;
  int b = idx >> 3, h = idx & 7;
  const int qo = h * HD, ko = HID + h * HD, vo = 2 * HID + h * HD;
  const float* base = qkv + (size_t)b * 3 * 768;
  const float sc = 0.17677669529663687f;
  for (int q = 0; q < 3; ++q) {
    float srow[3];
    for (int k = 0; k < 3; ++k) {
      float acc = 0.f;
      for (int d = 0; d < HD; ++d)
        acc = fmaf(base[q * 768 + qo + d], base[k * 768 + ko + d], acc);
      srow[k] = acc * sc;
    }
    float mx = fmaxf(srow[0], fmaxf(srow[1], srow[2]));
    float e0 = __expf(srow[0] - mx), e1 = __expf(srow[1] - mx), e2 = __expf(srow[2] - mx);
    float inv = 1.f / (e0 + e1 + e2);
    float a0 = e0 * inv, a1 = e1 * inv, a2 = e2 * inv;
    for (int d = 0; d < HD; ++d) {
      float o = a0 * base[0 * 768 + vo + d] + a1 * base[1 * 768 + vo + d] +
                a2 * base[2 * 768 + vo + d];
      aout[(size_t)(b * 3 + q) * HID + h * HD + d] = o;
    }
    atomicAdd(&attnw[b * 9 + q * 3 + 0], a0 * 0.125f);
    atomicAdd(&attnw[b * 9 + q * 3 + 1], a1 * 0.125f);
    atomicAdd(&attnw[b * 9 + q * 3 + 2], a2 * 0.125f);
  }
}

__global__ void mean3_kernel(const float* __restrict__ fused, float* __restrict__ o) {
  int i = blockIdx.x * blockDim.x + threadIdx.x;
  if (i >= NN * HID) return;
  int n = i >> 8, c = i & 255;
  o[i] = (fused[(size_t)(n * 3) * HID + c] + fused[(size_t)(n * 3 + 1) * HID + c] +
          fused[(size_t)(n * 3 + 2) * HID + c]) * (1.f / 3.f);
}

__global__ void unc_finalize_kernel(const float* __restrict__ u,
                                    float* mp, float* lv, float* uc) {
  int n = blockIdx.x * blockDim.x + threadIdx.x;
  if (n >= NN) return;
  float m = u[n * 2], l = u[n * 2 + 1];
  mp[n] = m; lv[n] = l; uc[n] = __expf(0.5f * l);
}

__global__ void pool_mean_kernel(const float* __restrict__ fin, float* pooled) {
  int c = threadIdx.x;  // 256
  float s = 0.f;
  for (int n = 0; n < NN; ++n) s += fin[(size_t)n * HID + c];
  pooled[c] = s * (1.f / NN);
}

__global__ void lr_head_kernel(const float* __restrict__ pooled,
                               const float* __restrict__ w1, const float* __restrict__ b1,
                               const float* __restrict__ w2, const float* __restrict__ b2,
                               float* outp) {
  __shared__ float h[64];
  int j = threadIdx.x;  // 64
  float acc = b1[j];
  for (int c = 0; c < HID; ++c) acc = fmaf(w1[j * HID + c], pooled[c], acc);
  h[j] = fmaxf(acc, 0.f);
  __syncthreads();
  if (j == 0) {
    float v = b2[0];
    for (int k = 0; k < 64; ++k) v = fmaf(w2[k], h[k], v);
    outp[0] = 1.f / (1.f + __expf(-v));
  }
}

__global__ void wts_finalize_kernel(const float* wacc, float* wts) {
  int t = threadIdx.x;
  if (t < 3) wts[t] = wacc[t] * (1.f / NN);
}

// ---------------------------------------------------------------------------
static inline void gemm(const float* A, const float* W, const float* b, float* C,
                        int M, int N, int K, int relu, hipStream_t st) {
  if ((M % 32 == 0) && (N % 64 == 0)) {
    dim3 g(N / 64, M / 32);
    gemm_wt_t24_kernel<<<g, 32, 0, st>>>(A, W, b, C, M, N, K, relu);
  } else {
    dim3 g((N + 15) / 16, (M + 15) / 16);
    gemm_wt_kernel<<<g, 32, 0, st>>>(A, W, b, C, M, N, K, relu);
  }
}

extern "C" void kernel_launch(void* const* d_in, const int* in_sizes, int n_in,
                              void* d_out, int out_size, void* d_ws, size_t ws_size,
                              hipStream_t stream) {
  if (n_in < 73) return;  // expects 3 tensor inputs + 70 param leaves (jax pytree order)
  const float* node_f = (const float*)d_in[0];
  const int*   ei     = (const int*)d_in[1];      // [2, NE]
  const float* ts     = (const float*)d_in[2];
  const int* src = ei, *dst = ei + NE;
  auto P = [&](int i) { return (const float*)d_in[i]; };
  // pytree (alphabetical dict keys): fusion(3..6: in_b,in_w,out_b,out_w),
  // layers s: 7+12s (+0 in_b,+1 in_w,+2 l1_b,+3 l1_w,+4 l2_b,+5 l2_w,
  //                  +6 ln1_b,+7 ln1_s,+8 ln2_b,+9 ln2_s,+10 out_b,+11 out_w),
  // lr 43..46 (b1,b2,w1,w2), node_proj 47,48 (b,w), out_proj 49,50 (b,w),
  // time_enc s: 51+6s (b1,b2,b3,w1,w2,w3), unc 69..72 (b1,b2,w1,w2)

  float* ws = (float*)d_ws;
  const size_t X = 0, CNT = X + (size_t)NN * HID, S64 = CNT + NN,
               TBUF = S64 + 3ull * NN * 64, H0 = TBUF + (size_t)NN * HID,
               QKV = H0 + (size_t)NN * HID, ATTN = QKV + (size_t)NN * 768,
               TMP1 = ATTN + (size_t)NN * HID, TMP2 = TMP1 + (size_t)NN * 512,
               EMB = TMP2 + (size_t)NN * HID, STACK = EMB + 3ull * NN * HID,
               FQKV = STACK + 3ull * NN * HID, FAOUT = FQKV + 3ull * NN * 768,
               FUSED = FAOUT + 3ull * NN * HID, MEAN3 = FUSED + 3ull * NN * HID,
               UNCH = MEAN3 + (size_t)NN * HID, UBUF = UNCH + (size_t)NN * 128,
               POOLED = UBUF + (size_t)NN * 2, WACC = POOLED + HID;

  float* outv = (float*)d_out;
  float* o_final = outv;                 // [2048,256]
  float* o_mp    = outv + 524288;        // [2048]
  float* o_lv    = outv + 526336;        // [2048]
  float* o_uc    = outv + 528384;        // [2048]
  float* o_plr   = outv + 530432;        // [1]
  float* o_wts   = outv + 530433;        // [3]
  float* o_aw    = outv + 530436;        // [2048,3,3]

  hipMemsetAsync(ws + CNT, 0, NN * sizeof(float), stream);
  hipMemsetAsync(ws + S64, 0, 3ull * NN * 64 * sizeof(float), stream);
  hipMemsetAsync(ws + WACC, 0, 3 * sizeof(float), stream);
  hipMemsetAsync(o_aw, 0, (size_t)NN * 9 * sizeof(float), stream);

  // x = node_proj
  gemm(node_f, P(48), P(47), ws + X, NN, HID, ND, 0, stream);
  edge_cnt_kernel<<<(NE + 255) / 256, 256, 0, stream>>>(src, dst, ws + CNT);

  const float inv_scale[3] = {1.f, 0.1f, 0.01f};
  for (int s = 0; s < 3; ++s) {
    const int T = 51 + 6 * s;
    edge_time_mlp_kernel<<<NE, 64, 0, stream>>>(
        ts, src, dst, P(T + 3), P(T + 0), P(T + 4), P(T + 1),
        ws + S64 + (size_t)s * NN * 64, inv_scale[s]);
  }

  for (int s = 0; s < 3; ++s) {
    const int T = 51 + 6 * s, L = 7 + 12 * s;
    // ntf = (segsum64 @ w3^T)/cnt + b3 ; h0 = x + ntf
    gemm(ws + S64 + (size_t)s * NN * 64, P(T + 5), nullptr, ws + TBUF, NN, HID, 64, 0, stream);
    ntf_combine_kernel<<<(NN * HID + 255) / 256, 256, 0, stream>>>(
        ws + X, ws + TBUF, ws + CNT, P(T + 2), ws + H0);
    // attention
    gemm(ws + H0, P(L + 1), P(L + 0), ws + QKV, NN, 3 * HID, HID, 0, stream);
    flash_attn_kernel<<<dim3(NN / 16, NH), 32, 0, stream>>>(ws + QKV, ws + ATTN);
    gemm(ws + ATTN, P(L + 11), P(L + 10), ws + TMP2, NN, HID, HID, 0, stream);
    add_ln_kernel<<<NN, 256, 0, stream>>>(ws + H0, ws + TMP2, P(L + 7), P(L + 6), ws + ATTN);
    // FFN
    gemm(ws + ATTN, P(L + 3), P(L + 2), ws + TMP1, NN, 2 * HID, HID, 1, stream);
    gemm(ws + TMP1, P(L + 5), P(L + 4), ws + TMP2, NN, HID, 2 * HID, 0, stream);
    add_ln_kernel<<<NN, 256, 0, stream>>>(ws + ATTN, ws + TMP2, P(L + 9), P(L + 8),
                                          ws + EMB + (size_t)s * NN * HID);
    rownorm_acc_kernel<<<NN, 256, 0, stream>>>(ws + EMB + (size_t)s * NN * HID, ws + WACC + s);
  }

  // fusion MHA over stacked [2048,3,256]
  interleave3_kernel<<<(NN * HID + 255) / 256, 256, 0, stream>>>(
      ws + EMB, ws + EMB + (size_t)NN * HID, ws + EMB + 2ull * NN * HID, ws + STACK);
  gemm(ws + STACK, P(4), P(3), ws + FQKV, 3 * NN, 3 * HID, HID, 0, stream);
  fusion_attn_kernel<<<(NN * NH + 127) / 128, 128, 0, stream>>>(ws + FQKV, ws + FAOUT, o_aw);
  gemm(ws + FAOUT, P(6), P(5), ws + FUSED, 3 * NN, HID, HID, 0, stream);
  mean3_kernel<<<(NN * HID + 255) / 256, 256, 0, stream>>>(ws + FUSED, ws + MEAN3);
  gemm(ws + MEAN3, P(50), P(49), o_final, NN, HID, HID, 0, stream);

  // uncertainty head
  gemm(o_final, P(71), P(69), ws + UNCH, NN, 128, HID, 1, stream);
  gemm(ws + UNCH, P(72), P(70), ws + UBUF, NN, 2, 128, 0, stream);
  unc_finalize_kernel<<<(NN + 255) / 256, 256, 0, stream>>>(ws + UBUF, o_mp, o_lv, o_uc);

  // lr head + wts
  pool_mean_kernel<<<1, 256, 0, stream>>>(o_final, ws + POOLED);
  lr_head_kernel<<<1, 64, 0, stream>>>(ws + POOLED, P(45), P(43), P(46), P(44), o_plr);
  wts_finalize_kernel<<<1, 32, 0, stream>>>(ws + WACC, o_wts);
}